// MetaRoPE_45810121179387
// MI455X (gfx1250) — compile-verified
//
#include <hip/hip_runtime.h>
#include <hip/hip_bf16.h>

// ---- problem dims (match reference: x [4,32,4096,64] f32, rotate_m [4096,64,64] f32) ----
#define B_DIM 4
#define H_DIM 32
#define BH    (B_DIM * H_DIM)      // 128
#define S_DIM 4096
#define D_DIM 64
#define HALF  (D_DIM / 2)          // 32 rotation pairs per position

#define CHUNK_S      32            // positions handled per block
#define BH_PER_BLK   4             // (b,h) slices handled per block
#define THREADS      256           // 8 wave32 waves

typedef float  f32x4 __attribute__((ext_vector_type(4)));
typedef float  f32x2 __attribute__((ext_vector_type(2)));
typedef unsigned int u32x4 __attribute__((ext_vector_type(4)));
typedef int    i32x8 __attribute__((ext_vector_type(8)));
typedef int    i32x4 __attribute__((ext_vector_type(4)));

// RoPE, bandwidth-bound formulation (~270 MB moved; ~11.5 us floor @ 23.3 TB/s).
//   - TDM gathers the (cos, -sin) pairs for CHUNK_S positions into LDS
//     (one 2-D descriptor per position so token_positions is honored).
//   - Coefficients read back with explicit ds_load_b64 (LDS path, not flat).
//   - Main loop streams x with NT 128-bit loads, rotates, NT 128-bit stores.
__global__ __launch_bounds__(THREADS)
void MetaRoPE_rope_tdm_kernel(const float* __restrict__ x,
                              const int*   __restrict__ token_positions,
                              const float* __restrict__ rotate_m,
                              float*       __restrict__ out) {
    __shared__ f32x2 coef[CHUNK_S * HALF];   // coef[sl*HALF + k] = (cos, -sin)  (8 KiB)

    const int s0 = blockIdx.x * CHUNK_S;
    // Low 32 bits of a generic LDS address are the LDS byte offset
    const unsigned ldsBase = (unsigned)(size_t)&coef[0];

    // ---------------- TDM: DMA the rotation coefficients into LDS ----------------
    // Pair k of position p lives at rotate_m + p*D*D + (2k*D + 2k) floats:
    // 8 contiguous bytes (c, -s), stride 520 B between pairs -> 65 x 8-byte elements.
    if (threadIdx.x < 32) {   // one wave issues; TDM ignores EXEC, so guard by branch
        // D# group 1: data_size=8B (code 3), tensor_dim0=1, tensor_dim1=32,
        //             tile_dim0=1, tile_dim1=32, tile_dim2=0, stride0=65, stride1=0
        const i32x8 g1 = { 0x00030000,      // mask=0, data_size=3(8B), no pad/iterate
                           (1  << 16),      // abar_addr=0 | tensor_dim0.lo16 = 1
                           (32 << 16),      // tensor_dim0.hi16=0 | tensor_dim1.lo16 = 32
                           (1  << 16),      // tensor_dim1.hi16=0 | tile_dim0 = 1
                           32,              // tile_dim1 = 32 | tile_dim2 = 0
                           65,              // tensor_dim0_stride.lo32 = 65 elements
                           0, 0 };
        const i32x4 g2 = { 0, 0, 0, 0 };    // 2-D tile: dims 2..4 unused
        const i32x4 g3 = { 0, 0, 0, 0 };
        const i32x8 g4 = { 0, 0, 0, 0, 0, 0, 0, 0 };   // unused trailing group (VADDR4)

        for (int sl = 0; sl < CHUNK_S; ++sl) {
            const int pos = token_positions[s0 + sl];                  // uniform -> SMEM
            const unsigned long long ga =
                (unsigned long long)rotate_m +
                (unsigned long long)pos * (D_DIM * D_DIM * 4ull);
            const u32x4 g0 = {
                1u,                                          // count=1, user mode
                ldsBase + (unsigned)(sl * (HALF * 8)),       // lds_addr (bytes)
                (unsigned)(ga & 0xffffffffull),              // global_addr[31:0]
                (unsigned)(((ga >> 32) & 0x01ffffffull)      // global_addr[56:32]
                           | 0x80000000u)                    // type = 2 ("image")
            };
            __builtin_amdgcn_tensor_load_to_lds(g0, g1, g2, g3, g4, 0);
        }
        __builtin_amdgcn_s_wait_tensorcnt(0);                // drain TDM in issuing wave
    }
    __syncthreads();                                         // publish LDS to all waves

    // ---------------- streaming rotate: NT 128-bit in / out ----------------
    const int tid = threadIdx.x;
    const int bh0 = blockIdx.y * BH_PER_BLK;

    #pragma unroll
    for (int p = 0; p < (CHUNK_S * D_DIM / 4) / THREADS; ++p) {       // 2 passes
        const int flat = p * THREADS + tid;      // float4 index in the chunk tile
        const int sl   = flat >> 4;              // position within chunk
        const int c4   = flat & 15;              // float4 column (d = 4*c4)

        // Explicit LDS loads of this thread's two (c, -s) pairs. The asm embeds
        // the DScnt wait because the compiler doesn't track asm memory results.
        const unsigned off0 = ldsBase + (unsigned)((sl * HALF + c4 * 2) * 8);
        f32x2 cs0, cs1;
        asm volatile("ds_load_b64 %0, %2\n\t"
                     "ds_load_b64 %1, %2 offset:8\n\t"
                     "s_wait_dscnt 0x0"
                     : "=&v"(cs0), "=&v"(cs1)
                     : "v"(off0));

        const size_t tileOff = (size_t)(s0 + sl) * D_DIM + (size_t)c4 * 4;

        #pragma unroll
        for (int j = 0; j < BH_PER_BLK; ++j) {
            const size_t off = ((size_t)(bh0 + j) * S_DIM) * D_DIM + tileOff;

            const f32x4 v = __builtin_nontemporal_load((const f32x4*)(x + off));
            f32x4 o;
            o.x = fmaf(cs0.x, v.x,  cs0.y * v.y);    // c*x0 - s*x1
            o.y = fmaf(cs0.x, v.y, -cs0.y * v.x);    // s*x0 + c*x1
            o.z = fmaf(cs1.x, v.z,  cs1.y * v.w);
            o.w = fmaf(cs1.x, v.w, -cs1.y * v.z);
            __builtin_nontemporal_store(o, (f32x4*)(out + off));
        }
    }
}

extern "C" void kernel_launch(void* const* d_in, const int* in_sizes, int n_in,
                              void* d_out, int out_size, void* d_ws, size_t ws_size,
                              hipStream_t stream) {
    (void)in_sizes; (void)n_in; (void)out_size; (void)d_ws; (void)ws_size;
    const float* x   = (const float*)d_in[0];
    const int*   tp  = (const int*)  d_in[1];
    const float* rot = (const float*)d_in[2];
    float*       out = (float*)d_out;

    dim3 grid(S_DIM / CHUNK_S, BH / BH_PER_BLK);   // 128 x 32 = 4096 blocks
    dim3 block(THREADS);
    MetaRoPE_rope_tdm_kernel<<<grid, block, 0, stream>>>(x, tp, rot, out);
}